// GNN_66494683676857
// MI455X (gfx1250) — compile-verified
//
#include <hip/hip_runtime.h>
#include <cstdint>
#include <cstddef>
#include <climits>

// ---------------------------------------------------------------------------
// Static problem config (matches reference)
// ---------------------------------------------------------------------------
#define G_     256
#define NP_    128
#define DEG_   8
#define FIN_   128
#define EMB_   64
#define HEADS_ 4
#define HC_    256           // HEADS*EMB
#define N0_    (G_ * NP_)    // 32768 nodes
#define E_     (G_ * NP_ * DEG_)  // 262144 edges
#define DENSE_ 256

typedef float v2f __attribute__((ext_vector_type(2)));
typedef float v8f __attribute__((ext_vector_type(8)));

static constexpr float kBnScale = 0.9999950000374997f; // 1/sqrt(1+1e-5)

// Order-preserving float<->int mapping so segment-max can use int atomicMax
// (deterministic, no dependence on float atomicMax support).
__device__ __forceinline__ int f2ord(float f) {
    int i = __float_as_int(f);
    return (i >= 0) ? i : (i ^ 0x7fffffff);
}
__device__ __forceinline__ float ord2f(int i) {
    return __int_as_float((i >= 0) ? i : (i ^ 0x7fffffff));
}

// ---------------------------------------------------------------------------
// Fill kernels
// ---------------------------------------------------------------------------
__global__ void k_fill_f32(float* __restrict__ p, float v, int n) {
    int i = blockIdx.x * blockDim.x + threadIdx.x;
    if (i < n) p[i] = v;
}
__global__ void k_fill_i32(int* __restrict__ p, int v, int n) {
    int i = blockIdx.x * blockDim.x + threadIdx.x;
    if (i < n) p[i] = v;
}

// ---------------------------------------------------------------------------
// WMMA f32 GEMM:  Y[M,N] = epilogue( X[M,K] @ W[K,N] + bias )
// One wave = 16x64 output strip (4 WMMA accumulators reusing the A fragment).
// Requires M%16==0, K%4==0, N%64==0 (true for all GEMMs routed here).
// Epilogue: +bias, optional ReLU, optional eval-BatchNorm (g/sqrt(1+eps), b).
//
// V_WMMA_F32_16X16X4_F32 operand layout (ISA 7.12.2):
//   A 16x4 : lane l -> row m=l&15; VGPR0=K=(l>>4)*2, VGPR1=K+1
//   B 4x16 : lane l -> col n=l&15; VGPR0=K=(l>>4)*2, VGPR1=K+1
//   C 16x16: VGPR r -> M=r (lanes 0-15), M=r+8 (lanes 16-31); N=lane&15
// ---------------------------------------------------------------------------
__global__ __launch_bounds__(128) void k_gemm_wmma(
    const float* __restrict__ X, const float* __restrict__ W,
    const float* __restrict__ bias, float* __restrict__ Y,
    int M, int K, int N, int doRelu,
    const float* __restrict__ bnG, const float* __restrict__ bnB)
{
    const int wave = threadIdx.x >> 5;
    const int lane = threadIdx.x & 31;
    const int lrow = lane & 15;
    const int lhi  = lane >> 4;
    const int n0   = blockIdx.x * 64;
    const int m0   = (blockIdx.y * 4 + wave) * 16;
    if (m0 >= M) return;                       // wave-uniform: EXEC stays all-1

    v8f acc0 = {}; v8f acc1 = {}; v8f acc2 = {}; v8f acc3 = {};
    const float* xp = X + (size_t)(m0 + lrow) * K + (lhi << 1);
    const int nc = n0 + lrow;
    for (int kk = 0; kk < K; kk += 4) {
        v2f a;
        a[0] = xp[kk];
        a[1] = xp[kk + 1];
        const float* wp = W + (size_t)(kk + (lhi << 1)) * N + nc;
        v2f b0, b1, b2, b3;
        b0[0] = wp[0];   b0[1] = wp[N];
        b1[0] = wp[16];  b1[1] = wp[N + 16];
        b2[0] = wp[32];  b2[1] = wp[N + 32];
        b3[0] = wp[48];  b3[1] = wp[N + 48];
        acc0 = __builtin_amdgcn_wmma_f32_16x16x4_f32(false, a, false, b0, (short)0, acc0, false, false);
        acc1 = __builtin_amdgcn_wmma_f32_16x16x4_f32(false, a, false, b1, (short)0, acc1, false, false);
        acc2 = __builtin_amdgcn_wmma_f32_16x16x4_f32(false, a, false, b2, (short)0, acc2, false, false);
        acc3 = __builtin_amdgcn_wmma_f32_16x16x4_f32(false, a, false, b3, (short)0, acc3, false, false);
    }
    v8f accs[4] = {acc0, acc1, acc2, acc3};
    #pragma unroll
    for (int t = 0; t < 4; ++t) {
        const int col = nc + t * 16;
        const float bv = bias ? bias[col] : 0.0f;
        const float sg = bnG ? (bnG[col] * kBnScale) : 1.0f;
        const float sb = bnB ? bnB[col] : 0.0f;
        #pragma unroll
        for (int r = 0; r < 8; ++r) {
            const int row = m0 + r + (lhi << 3);
            float vv = accs[t][r] + bv;
            if (doRelu) vv = fmaxf(vv, 0.0f);
            vv = vv * sg + sb;
            Y[(size_t)row * N + col] = vv;
        }
    }
}

// ---------------------------------------------------------------------------
// Edge attention: logits + segment max (order-preserving int atomicMax)
// ---------------------------------------------------------------------------
__global__ void k_edge_logits(const float* __restrict__ q, const float* __restrict__ k,
                              const int* __restrict__ srcP, const int* __restrict__ dstP,
                              const int* __restrict__ maskP,
                              float* __restrict__ logit, int* __restrict__ msegI, int Etot)
{
    int idx = blockIdx.x * blockDim.x + threadIdx.x;
    if (idx >= Etot * HEADS_) return;
    int e = idx >> 2, h = idx & 3;
    if (!maskP[e]) { logit[idx] = -1.0e30f; return; }
    int d = dstP[e], s = srcP[e];
    const float* qp = q + (size_t)d * HC_ + h * EMB_;
    const float* kp = k + (size_t)s * HC_ + h * EMB_;
    float sum = 0.f;
    #pragma unroll 8
    for (int c = 0; c < EMB_; ++c) sum += qp[c] * kp[c];
    sum *= 0.125f;                              // 1/sqrt(EMB)
    logit[idx] = sum;
    atomicMax(&msegI[d * HEADS_ + h], f2ord(sum));
}

// exp(logit - m[dst]) in place, segment-sum denominator
__global__ void k_edge_exp(float* __restrict__ pl, const int* __restrict__ msegI,
                           float* __restrict__ den,
                           const int* __restrict__ dstP, const int* __restrict__ maskP, int Etot)
{
    int idx = blockIdx.x * blockDim.x + threadIdx.x;
    if (idx >= Etot * HEADS_) return;
    int e = idx >> 2, h = idx & 3;
    if (!maskP[e]) { pl[idx] = 0.0f; return; }
    int d = dstP[e];
    float m = ord2f(msegI[d * HEADS_ + h]);
    float pe = expf(pl[idx] - m);
    pl[idx] = pe;
    atomicAdd(&den[d * HEADS_ + h], pe);
}

// out[dst,h,c] += (p/(den+eps)) * v[src,h,c]
__global__ void k_edge_aggr(const float* __restrict__ p, const float* __restrict__ den,
                            const float* __restrict__ v,
                            const int* __restrict__ srcP, const int* __restrict__ dstP,
                            const int* __restrict__ maskP,
                            float* __restrict__ outb, int Etot)
{
    long long idx = (long long)blockIdx.x * blockDim.x + threadIdx.x;
    if (idx >= (long long)Etot * HC_) return;
    int e = (int)(idx >> 8);
    int c = (int)(idx & 255);
    if (!maskP[e]) return;
    int h = c >> 6;
    int d = dstP[e];
    float coef = p[e * HEADS_ + h] / (den[d * HEADS_ + h] + 1e-16f);
    atomicAdd(&outb[(size_t)d * HC_ + c], coef * v[(size_t)srcP[e] * HC_ + c]);
}

// ---------------------------------------------------------------------------
// Beta gate: one wave32 per node; xc = beta*r + (1-beta)*out
// ---------------------------------------------------------------------------
__global__ __launch_bounds__(128) void k_combine(const float* __restrict__ outb,
                                                 const float* __restrict__ skipb,
                                                 const float* __restrict__ wb,
                                                 float* __restrict__ xc, int Nn)
{
    int node = blockIdx.x * 4 + (threadIdx.x >> 5);
    int lane = threadIdx.x & 31;
    if (node >= Nn) return;
    const float* o = outb + (size_t)node * HC_;
    const float* r = skipb + (size_t)node * HC_;
    float part = 0.f;
    for (int j = lane; j < HC_; j += 32) {
        float ov = o[j], rv = r[j];
        part += wb[j] * ov + wb[HC_ + j] * rv + wb[2 * HC_ + j] * (ov - rv);
    }
    #pragma unroll
    for (int m = 16; m >= 1; m >>= 1) part += __shfl_xor(part, m, 32);
    float beta = 1.f / (1.f + expf(-part));
    for (int j = lane; j < HC_; j += 32) {
        float ov = o[j], rv = r[j];
        xc[(size_t)node * HC_ + j] = beta * rv + (1.f - beta) * ov;
    }
}

// ---------------------------------------------------------------------------
// TopK pooling: one block per graph, thread per node; rank via O(n^2) count
// with JAX top_k tie-breaking (lower index wins on equal scores).
// ---------------------------------------------------------------------------
__global__ void k_pool(const float* __restrict__ x, const float* __restrict__ w,
                       float* __restrict__ xn, int* __restrict__ mapping,
                       int nper, int kk)
{
    __shared__ float s[128];
    int g = blockIdx.x, i = threadIdx.x;
    float wn = 0.f;
    #pragma unroll
    for (int c = 0; c < EMB_; ++c) wn += w[c] * w[c];
    wn = sqrtf(wn);
    const float* xr = x + ((size_t)g * nper + i) * EMB_;
    float dot = 0.f;
    #pragma unroll 8
    for (int c = 0; c < EMB_; ++c) dot += xr[c] * w[c];
    float sc = tanhf(dot / wn);
    s[i] = sc;
    __syncthreads();
    int rank = 0;
    for (int j = 0; j < nper; ++j) {
        float sj = s[j];
        rank += (sj > sc) || (sj == sc && j < i);
    }
    int nid = -1;
    if (rank < kk) {
        nid = g * kk + rank;
        float* dp = xn + (size_t)nid * EMB_;
        for (int c = 0; c < EMB_; ++c) dp[c] = xr[c] * sc;
    }
    mapping[g * nper + i] = nid;
}

__global__ void k_remap(const int* __restrict__ so, const int* __restrict__ dO,
                        const int* __restrict__ mo, const int* __restrict__ mapping,
                        int* __restrict__ sn, int* __restrict__ dn, int* __restrict__ mn, int Etot)
{
    int e = blockIdx.x * blockDim.x + threadIdx.x;
    if (e >= Etot) return;
    int v = mo[e];
    int ns = mapping[so[e]], nd = mapping[dO[e]];
    v = v && (ns >= 0) && (nd >= 0);
    sn[e] = v ? ns : 0;
    dn[e] = v ? nd : 0;
    mn[e] = v;
}

// Graph readout: hrep[g, 0:64] += max, hrep[g, 64:128] += mean
__global__ void k_graph_rep(const float* __restrict__ xn, float* __restrict__ hrep, int nper)
{
    int g = blockIdx.x, c = threadIdx.x;   // 64 threads
    float mx = -3.0e38f, sm = 0.f;
    const float* base = xn + (size_t)g * nper * EMB_ + c;
    for (int n = 0; n < nper; ++n) {
        float v = base[(size_t)n * EMB_];
        mx = fmaxf(mx, v);
        sm += v;
    }
    hrep[g * 2 * EMB_ + c]        += mx;
    hrep[g * 2 * EMB_ + EMB_ + c] += sm / (float)nper;
}

// Final tiny GEMM: [G,128] @ [128,3] + b -> d_out
__global__ void k_lin3(const float* __restrict__ h, const float* __restrict__ w,
                       const float* __restrict__ b, float* __restrict__ outp, int Gn)
{
    int idx = blockIdx.x * blockDim.x + threadIdx.x;
    if (idx >= Gn * 3) return;
    int g = idx / 3, o = idx - g * 3;
    float s = b[o];
    #pragma unroll 8
    for (int j = 0; j < 2 * EMB_; ++j) s += h[g * 2 * EMB_ + j] * w[j * 3 + o];
    outp[idx] = s;
}

// ---------------------------------------------------------------------------
// Input-leaf index map.
// d_in[0..2] = x, edge_index, batch_index (dict insertion order).
// params leaves in JAX pytree order (dict keys sorted recursively):
//  bn1{b,g}, conv1{beta.w, k.b,k.w, q.b,q.w, skip.b,skip.w, v.b,v.w},
//  layers[i]{bn{b,g}, conv{...}, pool_w, transf{b,w}},
//  lin1{b,w}, lin2{b,w}, lin3{b,w}, transf1{b,w}
// ---------------------------------------------------------------------------
enum {
    IN_X = 0, IN_EI = 1, IN_BATCH = 2,
    P_BN1_B = 3, P_BN1_G = 4,
    P_C1_BETA = 5, P_C1_KB = 6, P_C1_KW = 7, P_C1_QB = 8, P_C1_QW = 9,
    P_C1_SB = 10, P_C1_SW = 11, P_C1_VB = 12, P_C1_VW = 13,
    P_L0 = 14,   // each layer is 14 leaves
    P_LIN1_B = 56, P_LIN1_W = 57, P_LIN2_B = 58, P_LIN2_W = 59,
    P_LIN3_B = 60, P_LIN3_W = 61, P_TR1_B = 62, P_TR1_W = 63,
};
enum { L_BN_B = 0, L_BN_G = 1, L_BETA = 2, L_KB = 3, L_KW = 4, L_QB = 5, L_QW = 6,
       L_SB = 7, L_SW = 8, L_VB = 9, L_VW = 10, L_POOLW = 11, L_TRB = 12, L_TRW = 13 };

extern "C" void kernel_launch(void* const* d_in, const int* in_sizes, int n_in,
                              void* d_out, int out_size, void* d_ws, size_t ws_size,
                              hipStream_t stream)
{
    (void)in_sizes; (void)n_in; (void)out_size; (void)ws_size;

    const float* x_in = (const float*)d_in[IN_X];
    const int*   ei   = (const int*)d_in[IN_EI];
    const int*   src0 = ei;
    const int*   dst0 = ei + E_;
    auto F = [&](int i) { return (const float*)d_in[i]; };

    // ---- workspace arena (~162 MB, fully re-initialized every call) ----
    float* ws = (float*)d_ws;
    size_t off = 0;
    auto A = [&](size_t n) { float* p = ws + off; off += n; return p; };
    float* bufQ = A((size_t)N0_ * HC_);       // q; later re-used as attention 'out'
    float* bufK = A((size_t)N0_ * HC_);       // k; later re-used as combined x
    float* bufV = A((size_t)N0_ * HC_);       // v
    float* bufS = A((size_t)N0_ * HC_);       // skip
    float* xA   = A((size_t)N0_ * EMB_);
    float* xB   = A((size_t)N0_ * EMB_);
    float* plog = A((size_t)E_ * HEADS_);     // logits -> p (in place)
    int*   msegI = (int*)A((size_t)N0_ * HEADS_);
    float* den  = A((size_t)N0_ * HEADS_);
    int*   mapping = (int*)A((size_t)N0_);
    int*   srcB = (int*)A(E_); int* dstB = (int*)A(E_); int* maskB = (int*)A(E_);
    int*   srcC = (int*)A(E_); int* dstC = (int*)A(E_); int* maskC = (int*)A(E_);
    int*   maskA = (int*)A(E_);
    float* hrep = A((size_t)G_ * 2 * EMB_);
    float* mlp1 = A((size_t)G_ * DENSE_);
    float* mlp2 = A((size_t)G_ * (DENSE_ / 2));

    auto fillF = [&](float* p, float v, int n) {
        k_fill_f32<<<(n + 255) / 256, 256, 0, stream>>>(p, v, n);
    };
    auto fillI = [&](int* p, int v, int n) {
        k_fill_i32<<<(n + 255) / 256, 256, 0, stream>>>(p, v, n);
    };
    auto gemm = [&](const float* X, const float* W, const float* B, float* Y,
                    int M, int K, int N, int relu, const float* g, const float* b) {
        dim3 grid(N / 64, (M + 63) / 64);
        k_gemm_wmma<<<grid, 128, 0, stream>>>(X, W, B, Y, M, K, N, relu, g, b);
    };

    // TransformerConv (beta=True): produces combined HC-wide features in bufK
    auto tconv = [&](const float* xin, int K, int Nn,
                     const int* sP, const int* dP, const int* mP,
                     const float* qw, const float* qb, const float* kw, const float* kb,
                     const float* vw, const float* vb, const float* sw, const float* sb,
                     const float* betaw) {
        gemm(xin, qw, qb, bufQ, Nn, K, HC_, 0, nullptr, nullptr);
        gemm(xin, kw, kb, bufK, Nn, K, HC_, 0, nullptr, nullptr);
        gemm(xin, vw, vb, bufV, Nn, K, HC_, 0, nullptr, nullptr);
        gemm(xin, sw, sb, bufS, Nn, K, HC_, 0, nullptr, nullptr);
        fillI(msegI, INT_MIN, Nn * HEADS_);
        fillF(den, 0.f, Nn * HEADS_);
        k_edge_logits<<<(E_ * HEADS_ + 255) / 256, 256, 0, stream>>>(
            bufQ, bufK, sP, dP, mP, plog, msegI, E_);
        fillF(bufQ, 0.f, Nn * HC_);                     // bufQ becomes 'out'
        k_edge_exp<<<(E_ * HEADS_ + 255) / 256, 256, 0, stream>>>(
            plog, msegI, den, dP, mP, E_);
        k_edge_aggr<<<(unsigned)(((long long)E_ * HC_ + 255) / 256), 256, 0, stream>>>(
            plog, den, bufV, sP, dP, mP, bufQ, E_);
        k_combine<<<(Nn + 3) / 4, 128, 0, stream>>>(bufQ, bufS, betaw, bufK, Nn);
    };

    // ---- init persistent accumulators / masks ----
    fillI(maskA, 1, E_);
    fillF(hrep, 0.f, G_ * 2 * EMB_);

    // ---- stage 1: conv1 (F_IN -> HC) + transf1 + bn1 -> xA [N0, EMB] ----
    tconv(x_in, FIN_, N0_, src0, dst0, maskA,
          F(P_C1_QW), F(P_C1_QB), F(P_C1_KW), F(P_C1_KB),
          F(P_C1_VW), F(P_C1_VB), F(P_C1_SW), F(P_C1_SB), F(P_C1_BETA));
    gemm(bufK, F(P_TR1_W), F(P_TR1_B), xA, N0_, HC_, EMB_, 1, F(P_BN1_G), F(P_BN1_B));

    // ---- 3 layers: tconv -> transf+bn -> topk pool -> readout ----
    int nper = NP_;
    int Nn = N0_;
    const int* curSrc = src0; const int* curDst = dst0; const int* curMask = maskA;
    for (int l = 0; l < 3; ++l) {
        const int base = P_L0 + l * 14;
        tconv(xA, EMB_, Nn, curSrc, curDst, curMask,
              F(base + L_QW), F(base + L_QB), F(base + L_KW), F(base + L_KB),
              F(base + L_VW), F(base + L_VB), F(base + L_SW), F(base + L_SB),
              F(base + L_BETA));
        gemm(bufK, F(base + L_TRW), F(base + L_TRB), xB, Nn, HC_, EMB_, 1,
             F(base + L_BN_G), F(base + L_BN_B));

        const int kk = nper / 2;
        k_pool<<<G_, nper, 0, stream>>>(xB, F(base + L_POOLW), xA, mapping, nper, kk);
        if (l == 0) {
            k_remap<<<(E_ + 255) / 256, 256, 0, stream>>>(
                curSrc, curDst, curMask, mapping, srcB, dstB, maskB, E_);
            curSrc = srcB; curDst = dstB; curMask = maskB;
        } else if (l == 1) {
            k_remap<<<(E_ + 255) / 256, 256, 0, stream>>>(
                curSrc, curDst, curMask, mapping, srcC, dstC, maskC, E_);
            curSrc = srcC; curDst = dstC; curMask = maskC;
        } // after layer 2, edges are never consumed again
        nper = kk;
        Nn = G_ * nper;
        k_graph_rep<<<G_, EMB_, 0, stream>>>(xA, hrep, nper);
    }

    // ---- MLP head ----
    gemm(hrep, F(P_LIN1_W), F(P_LIN1_B), mlp1, G_, 2 * EMB_, DENSE_, 1, nullptr, nullptr);
    gemm(mlp1, F(P_LIN2_W), F(P_LIN2_B), mlp2, G_, DENSE_, DENSE_ / 2, 1, nullptr, nullptr);
    k_lin3<<<(G_ * 3 + 255) / 256, 256, 0, stream>>>(
        mlp2, F(P_LIN3_W), F(P_LIN3_B), (float*)d_out, G_);
}